// LogActivationLayer_12936441496124
// MI455X (gfx1250) — compile-verified
//
#include <hip/hip_runtime.h>
#include <hip/hip_bf16.h>

typedef __attribute__((ext_vector_type(2))) float v2f;
typedef __attribute__((ext_vector_type(8))) float v8f;

#define BATCH   8192
#define IN      64
#define OUT     64
#define NSPL    8
#define NBRK    20
#define NINT    19
#define BM      32          // batch rows per block

// Workspace layout (floats):
//   [0,      4096)  T1 = b1*gamma/64   layout [i][o]
//   [4096,   8192)  T2 = b2            layout [i][o]
//   [8192,  12288)  T3 = b3            layout [i][o]
//   [12288, 16384)  T4 = b4            layout [i][o]
//   [16384, 32768)  Wp: poly weights [k][o], k = p*64+i, value b_{5+p}*gamma/64
// total 128 KB

__global__ __launch_bounds__(256)
void la_prep_kernel(const float* __restrict__ raw_gamma,
                    const float* __restrict__ w,
                    const float* __restrict__ breaks,
                    const float* __restrict__ coefs,
                    const float* __restrict__ mu_p,
                    const float* __restrict__ sigma_p,
                    float* __restrict__ ws)
{
    int t = blockIdx.x * 256 + threadIdx.x;      // one thread per (o,i)
    if (t >= OUT * IN) return;
    int o = t / IN;
    int i = t - o * IN;

    float mu = mu_p[0], sigma = sigma_p[0];
    float wc = fminf(fmaxf(w[t], -5.5f), 37.9f);
    float wn = (wc - mu) / sigma;

    float b[NSPL];
#pragma unroll
    for (int s = 0; s < NSPL; ++s) {
        const float* br = breaks + s * NBRK;
        float wl = fminf(fmaxf(wn, br[0]), br[NBRK - 1] - 1e-6f);
        int idx = 0;
#pragma unroll
        for (int j = 1; j < NINT; ++j) idx += (wl >= br[j]) ? 1 : 0;
        const float* cf = coefs + (s * NINT + idx) * 4;
        float tt = wl - br[idx];
        b[s] = ((cf[0] * tt + cf[1]) * tt + cf[2]) * tt + cf[3];
    }

    float g     = raw_gamma[t];
    float gamma = __logf(1.0f + __expf(g));      // softplus
    float scale = gamma * (1.0f / OUT);

    float* T1 = ws;
    float* T2 = ws + 4096;
    float* T3 = ws + 8192;
    float* T4 = ws + 12288;
    float* Wp = ws + 16384;

    int io = i * OUT + o;                         // transposed for coalesced reads
    T1[io] = b[0] * scale;
    T2[io] = b[1];
    T3[io] = b[2];
    T4[io] = b[3];
#pragma unroll
    for (int p = 0; p < 4; ++p)
        Wp[(p * IN + i) * OUT + o] = b[4 + p] * scale;
}

__global__ __launch_bounds__(256)
void la_main_kernel(const float* __restrict__ x,
                    const float* __restrict__ ws,
                    float* __restrict__ out)
{
    __shared__ float xs[BM * IN];                 // 8 KB relu'd x tile

    const int row0 = blockIdx.x * BM;

    // cooperative load + relu of the 32x64 x tile
#pragma unroll
    for (int j = 0; j < (BM * IN) / 256; ++j) {
        int idx = threadIdx.x + j * 256;
        xs[idx] = fmaxf(x[row0 * IN + idx], 0.0f);
    }
    __syncthreads();

    const float* T1 = ws;
    const float* T2 = ws + 4096;
    const float* T3 = ws + 8192;
    const float* T4 = ws + 12288;
    const float* Wp = ws + 16384;

    const int wave  = threadIdx.x >> 5;           // 8 waves
    const int lane  = threadIdx.x & 31;
    const int hi    = lane >> 4;                  // 0/1 half-wave
    const int nl    = lane & 15;
    const int mtile = (wave >> 2) * 16;           // 0 or 16 (M tile)
    const int n     = (wave & 3) * 16 + nl;       // global output column

    // Accumulator in WMMA f32 16x16 C layout: vgpr v, lane -> M = mtile+v+8*hi, N = n
    v8f acc  = {0.f,0.f,0.f,0.f,0.f,0.f,0.f,0.f};
    v8f acc2 = {0.f,0.f,0.f,0.f,0.f,0.f,0.f,0.f};

    // ---- transcendental part: acc[v] += b1g * log1p(b2*log1p((e^{b3 x}-1)^{b4}))
    for (int i = 0; i < IN; ++i) {
        float t1 = T1[i * OUT + n];
        float t2 = T2[i * OUT + n];
        float t3 = T3[i * OUT + n];
        float t4 = T4[i * OUT + n];
#pragma unroll
        for (int v = 0; v < 8; ++v) {
            int   m  = mtile + v + 8 * hi;
            float xv = xs[m * IN + i];
            float u  = __expf(t3 * xv) - 1.0f;    // >= 0 (x>=0, b3>0)
            float bl = __expf(t4 * __logf(u));    // u^b4 ; u==0 -> exp(-inf)=0
            float l1 = __logf(1.0f + bl);
            float l2 = __logf(1.0f + t2 * l1);
            acc[v] += t1 * l2;
        }
    }

    // ---- polynomial part via chained V_WMMA_F32_16X16X4_F32
    // A: 16x4 f32 (x powers from LDS), B: 4x16 f32 (folded weights from L2)
    // A/B frag layout: vgpr0 = K {0 | 2}, vgpr1 = K {1 | 3} per half-wave
    const int am = mtile + nl;                    // A-matrix row for this lane
#pragma unroll
    for (int p = 0; p < 4; ++p) {                 // power index: x^(p+1)
#pragma unroll
        for (int c = 0; c < 16; ++c) {            // 16 K-chunks of 4 per power
            int i0 = c * 4 + hi * 2;
            float xa = xs[am * IN + i0];
            float xb = xs[am * IN + i0 + 1];
            float pa = xa, pb = xb;
#pragma unroll
            for (int q = 0; q < 3; ++q) {
                if (q < p) { pa *= xa; pb *= xb; }
            }
            v2f av = {pa, pb};
            v2f bv = {Wp[(p * IN + i0) * OUT + n],
                      Wp[(p * IN + i0 + 1) * OUT + n]};
            if (p < 2) {
                acc  = __builtin_amdgcn_wmma_f32_16x16x4_f32(
                           false, av, false, bv, (short)0, acc,  false, false);
            } else {
                acc2 = __builtin_amdgcn_wmma_f32_16x16x4_f32(
                           false, av, false, bv, (short)0, acc2, false, false);
            }
        }
    }

    // ---- store D (C-layout) to out (8192 x 64)
#pragma unroll
    for (int v = 0; v < 8; ++v) {
        int m = mtile + v + 8 * hi;
        out[(row0 + m) * OUT + n] = acc[v] + acc2[v];
    }
}

extern "C" void kernel_launch(void* const* d_in, const int* in_sizes, int n_in,
                              void* d_out, int out_size, void* d_ws, size_t ws_size,
                              hipStream_t stream) {
    const float* x         = (const float*)d_in[0];
    const float* raw_gamma = (const float*)d_in[1];
    const float* w         = (const float*)d_in[2];
    const float* breaks    = (const float*)d_in[3];
    const float* coefs     = (const float*)d_in[4];
    const float* mu        = (const float*)d_in[5];
    const float* sigma     = (const float*)d_in[6];
    float* out = (float*)d_out;
    float* ws  = (float*)d_ws;

    la_prep_kernel<<<(OUT * IN + 255) / 256, 256, 0, stream>>>(
        raw_gamma, w, breaks, coefs, mu, sigma, ws);
    la_main_kernel<<<BATCH / BM, 256, 0, stream>>>(x, ws, out);
}